// GAT_72645076845141
// MI455X (gfx1250) — compile-verified
//
#include <hip/hip_runtime.h>
#include <hip/hip_bf16.h>

typedef __attribute__((ext_vector_type(2))) float v2f;
typedef __attribute__((ext_vector_type(8))) float v8f;
typedef __attribute__((ext_vector_type(4))) unsigned int u32x4;
typedef __attribute__((ext_vector_type(4))) int i32x4;
typedef __attribute__((ext_vector_type(8))) int i32x8;

constexpr int   kN     = 4096;
constexpr int   kF     = 512;   // NFEAT
constexpr int   kD     = 64;    // NHID
constexpr int   kHeads = 8;
constexpr int   kC     = 41;    // NCLASS
constexpr float kAlpha = 0.2f;

// ---------------------------------------------------------------------------
// Tensor Data Mover: load a fully-in-bounds 2D f32 tile (tile==tensor extent)
// Global -> LDS.  D# per CDNA5 ISA.
//   group0: count=1 | lds_addr | global_addr[56:0] | type=2
//   group1: data_size=2(4B), dims/strides; no mask/pad/iterate
//   groups2/3: zero (2D tensor)
// ---------------------------------------------------------------------------
__device__ __forceinline__ void tdm_load_tile_f32(const float* gsrc,
                                                  unsigned lds_byte_off,
                                                  unsigned dim0,          // row length (elems)
                                                  unsigned dim1,          // #rows
                                                  unsigned dim0_stride) { // elems
  const unsigned long long ga = (unsigned long long)gsrc;
  u32x4 g0;
  g0[0] = 1u;                                            // count=1, user descriptor
  g0[1] = lds_byte_off;                                  // lds_addr (bytes)
  g0[2] = (unsigned)(ga & 0xFFFFFFFFull);                // global_addr[31:0]
  g0[3] = (unsigned)((ga >> 32) & 0x01FFFFFFull)         // global_addr[56:32]
          | (2u << 30);                                  // type=2 ("image")
  i32x8 g1;
  g1[0] = (int)(2u << 16);                               // data_size=2 (4B); mask/flags=0
  g1[1] = (int)((dim0 & 0xFFFFu) << 16);                 // tensor_dim0[15:0]  (bits 63:48)
  g1[2] = (int)((dim0 >> 16) | ((dim1 & 0xFFFFu) << 16));// tensor_dim0[31:16], tensor_dim1[15:0]
  g1[3] = (int)((dim1 >> 16) | ((dim0 & 0xFFFFu) << 16));// tensor_dim1[31:16], tile_dim0
  g1[4] = (int)(dim1 & 0xFFFFu);                         // tile_dim1; tile_dim2=0
  g1[5] = (int)dim0_stride;                              // tensor_dim0_stride[31:0]
  g1[6] = 0;                                             // stride0[47:32], stride1[15:0]
  g1[7] = 0;                                             // stride1[47:16]
  const i32x4 z4 = {0, 0, 0, 0};
#if __clang_major__ >= 23
  const i32x8 z8 = {0, 0, 0, 0, 0, 0, 0, 0};
  __builtin_amdgcn_tensor_load_to_lds(g0, g1, z4, z4, z8, 0);
#else
  __builtin_amdgcn_tensor_load_to_lds(g0, g1, z4, z4, 0);
#endif
}

__device__ __forceinline__ unsigned lds_off_of(const void* p) {
  // flat LDS pointer: aperture in high 32 bits, LDS byte offset in low 32
  return (unsigned)(unsigned long long)p;
}

// ---------------------------------------------------------------------------
// K1: Wh[h] = x @ W_heads[h]   (4096x512 @ 512x64) per head, f32 WMMA 16x16x4
// One wave per (16-row tile, head). A-tile (16x512 f32, 32KB) staged to LDS
// by the TDM; B streamed from global/L2.
// ---------------------------------------------------------------------------
__global__ __launch_bounds__(32) void k_gemm_heads(const float* __restrict__ x,
                                                   const float* __restrict__ Wheads,
                                                   float* __restrict__ Wh) {
  __shared__ float lds_x[16 * kF];   // 32 KB

  const int m0   = blockIdx.x * 16;
  const int h    = blockIdx.y;
  const int lane = threadIdx.x & 31;
  const int lo   = lane & 15;
  const int hi   = lane >> 4;
  const int kk   = hi << 1;   // K sub-offset: lanes 0-15 -> k+0/1, lanes 16-31 -> k+2/3

  tdm_load_tile_f32(x + (size_t)m0 * kF, lds_off_of(lds_x), kF, 16, kF);

  const float* W    = Wheads + (size_t)h * kF * kD;
  const float* arow = lds_x + (size_t)lo * kF;

  __builtin_amdgcn_s_wait_tensorcnt(0);   // A tile resident in LDS

  v8f a0 = {}, a1 = {}, a2 = {}, a3 = {};
  for (int k = 0; k < kF; k += 4) {
    v2f A;
    A.x = arow[k + kk];
    A.y = arow[k + kk + 1];
    const float* wp = W + (size_t)(k + kk) * kD + lo;
    v2f B0, B1, B2, B3;
    B0.x = wp[0];  B0.y = wp[kD + 0];
    B1.x = wp[16]; B1.y = wp[kD + 16];
    B2.x = wp[32]; B2.y = wp[kD + 32];
    B3.x = wp[48]; B3.y = wp[kD + 48];
    a0 = __builtin_amdgcn_wmma_f32_16x16x4_f32(false, A, false, B0, (short)0, a0, false, false);
    a1 = __builtin_amdgcn_wmma_f32_16x16x4_f32(false, A, false, B1, (short)0, a1, false, false);
    a2 = __builtin_amdgcn_wmma_f32_16x16x4_f32(false, A, false, B2, (short)0, a2, false, false);
    a3 = __builtin_amdgcn_wmma_f32_16x16x4_f32(false, A, false, B3, (short)0, a3, false, false);
  }
  float* out = Wh + ((size_t)h * kN + m0) * kD;
#pragma unroll
  for (int v = 0; v < 8; ++v) {
    const int m = v + (hi << 3);
    float* orow = out + (size_t)m * kD + lo;
    orow[0]  = a0[v];
    orow[16] = a1[v];
    orow[32] = a2[v];
    orow[48] = a3[v];
  }
}

// ---------------------------------------------------------------------------
// K2: s1[h,n] = Wh[h,n,:]·a1[h],  s2[h,n] = Wh[h,n,:]·a2[h]
// ---------------------------------------------------------------------------
__global__ void k_svec(const float* __restrict__ Wh, const float* __restrict__ a_heads,
                       float* __restrict__ s1, float* __restrict__ s2) {
  const int idx = blockIdx.x * blockDim.x + threadIdx.x;  // h*kN + n
  if (idx >= kHeads * kN) return;
  const int h = idx / kN;
  const float* row = Wh + (size_t)idx * kD;
  const float* a1  = a_heads + (size_t)h * 2 * kD;
  const float* a2  = a1 + kD;
  float x1 = 0.f, x2 = 0.f;
#pragma unroll 8
  for (int d = 0; d < kD; ++d) {
    const float w = row[d];
    x1 += w * a1[d];
    x2 += w * a2[d];
  }
  s1[idx] = x1;
  s2[idx] = x2;
}

// ---------------------------------------------------------------------------
// Block reductions over 256 threads
// ---------------------------------------------------------------------------
__device__ __forceinline__ float block_max(float v, float* red) {
  const int tid = threadIdx.x;
  red[tid] = v; __syncthreads();
#pragma unroll
  for (int s = 128; s > 0; s >>= 1) {
    if (tid < s) red[tid] = fmaxf(red[tid], red[tid + s]);
    __syncthreads();
  }
  const float r = red[0]; __syncthreads();
  return r;
}
__device__ __forceinline__ float block_sum(float v, float* red) {
  const int tid = threadIdx.x;
  red[tid] = v; __syncthreads();
#pragma unroll
  for (int s = 128; s > 0; s >>= 1) {
    if (tid < s) red[tid] += red[tid + s];
    __syncthreads();
  }
  const float r = red[0]; __syncthreads();
  return r;
}

// ---------------------------------------------------------------------------
// K3: layer-1 attention. One block per row i; adjacency nonzeros compacted
// into LDS once and reused for all 8 heads (adj read exactly once total).
// Masked entries contribute exp(NEG-max)==0 exactly, so skipping them is
// bit-equivalent to the reference softmax.
// ---------------------------------------------------------------------------
__global__ __launch_bounds__(256) void k_attn1(const float* __restrict__ adj,
                                               const float* __restrict__ Wh,
                                               const float* __restrict__ s1,
                                               const float* __restrict__ s2,
                                               float* __restrict__ hcat) {
  __shared__ int   idxs[kN];
  __shared__ float wv[kN];
  __shared__ float red[256];
  __shared__ int   cnt_s;

  const int i   = blockIdx.x;
  const int tid = threadIdx.x;
  if (tid == 0) cnt_s = 0;
  __syncthreads();

  const float* arow = adj + (size_t)i * kN;
  __builtin_prefetch(arow, 0, 1);  // global_prefetch_b8
  for (int j = tid; j < kN; j += 256) {
    if (arow[j] > 0.f) {
      const int p = atomicAdd(&cnt_s, 1);
      idxs[p] = j;
    }
  }
  __syncthreads();
  const int cnt = cnt_s;

  for (int h = 0; h < kHeads; ++h) {
    const float  s1i = s1[h * kN + i];
    const float* s2h = s2 + (size_t)h * kN;

    float lmax = -3.0e38f;
    for (int p = tid; p < cnt; p += 256) {
      float e = s1i + s2h[idxs[p]];
      e = e > 0.f ? e : kAlpha * e;
      wv[p] = e;
      lmax  = fmaxf(lmax, e);
    }
    const float mx = block_max(lmax, red);  // barrier also publishes wv[]

    float lsum = 0.f;
    for (int p = tid; p < cnt; p += 256) {
      const float w = __expf(wv[p] - mx);
      wv[p] = w;
      lsum += w;
    }
    const float invZ = 1.f / block_sum(lsum, red);

    const float* WhH = Wh + (size_t)h * kN * kD;
    if (tid < kD) {
      float acc = 0.f;
      for (int p = 0; p < cnt; ++p)
        acc += wv[p] * WhH[(size_t)idxs[p] * kD + tid];
      acc *= invZ;
      const float o = acc > 0.f ? acc : (__expf(acc) - 1.f);  // ELU
      hcat[(size_t)i * (kHeads * kD) + h * kD + tid] = o;
    }
    __syncthreads();  // protect wv/red/idxs reuse for next head
  }
}

// ---------------------------------------------------------------------------
// K4: Wh2 = hcat @ W_out   (4096x512 @ 512x41), f32 WMMA, N padded to 48.
// A-tile staged to LDS via TDM.
// ---------------------------------------------------------------------------
__global__ __launch_bounds__(32) void k_gemm_out(const float* __restrict__ hcat,
                                                 const float* __restrict__ W_out,
                                                 float* __restrict__ Wh2) {
  __shared__ float lds_a[16 * (kHeads * kD)];   // 32 KB

  const int m0   = blockIdx.x * 16;
  const int lane = threadIdx.x & 31;
  const int lo   = lane & 15;
  const int hi   = lane >> 4;
  const int kk   = hi << 1;

  tdm_load_tile_f32(hcat + (size_t)m0 * (kHeads * kD), lds_off_of(lds_a),
                    kHeads * kD, 16, kHeads * kD);

  const float* arow = lds_a + (size_t)lo * (kHeads * kD);

  __builtin_amdgcn_s_wait_tensorcnt(0);

  v8f a0 = {}, a1 = {}, a2 = {};
  for (int k = 0; k < kHeads * kD; k += 4) {
    v2f A;
    A.x = arow[k + kk];
    A.y = arow[k + kk + 1];
    const float* wp0 = W_out + (size_t)(k + kk) * kC;
    const float* wp1 = wp0 + kC;
    v2f B0, B1, B2;
    const int c0 = lo, c1 = 16 + lo, c2 = 32 + lo;
    B0.x = wp0[c0];                    B0.y = wp1[c0];
    B1.x = wp0[c1];                    B1.y = wp1[c1];
    B2.x = (c2 < kC) ? wp0[c2] : 0.f;  B2.y = (c2 < kC) ? wp1[c2] : 0.f;
    a0 = __builtin_amdgcn_wmma_f32_16x16x4_f32(false, A, false, B0, (short)0, a0, false, false);
    a1 = __builtin_amdgcn_wmma_f32_16x16x4_f32(false, A, false, B1, (short)0, a1, false, false);
    a2 = __builtin_amdgcn_wmma_f32_16x16x4_f32(false, A, false, B2, (short)0, a2, false, false);
  }
#pragma unroll
  for (int v = 0; v < 8; ++v) {
    const int m = v + (hi << 3);
    float* orow = Wh2 + (size_t)(m0 + m) * kC;
    orow[lo]      = a0[v];
    orow[16 + lo] = a1[v];
    if (32 + lo < kC) orow[32 + lo] = a2[v];
  }
}

// ---------------------------------------------------------------------------
// K5: t1[i] = Wh2[i]·a_out[:41],  t2[i] = Wh2[i]·a_out[41:]
// ---------------------------------------------------------------------------
__global__ void k_tvec(const float* __restrict__ Wh2, const float* __restrict__ a_out,
                       float* __restrict__ t1, float* __restrict__ t2) {
  const int i = blockIdx.x * blockDim.x + threadIdx.x;
  if (i >= kN) return;
  const float* row = Wh2 + (size_t)i * kC;
  float x1 = 0.f, x2 = 0.f;
  for (int c = 0; c < kC; ++c) {
    const float w = row[c];
    x1 += w * a_out[c];
    x2 += w * a_out[kC + c];
  }
  t1[i] = x1;
  t2[i] = x2;
}

// ---------------------------------------------------------------------------
// K6: layer-2 attention + ELU + fused log_softmax over 41 classes
// ---------------------------------------------------------------------------
__global__ __launch_bounds__(256) void k_attn2(const float* __restrict__ adj,
                                               const float* __restrict__ Wh2,
                                               const float* __restrict__ t1,
                                               const float* __restrict__ t2,
                                               float* __restrict__ out) {
  __shared__ int   idxs[kN];
  __shared__ float wv[kN];
  __shared__ float red[256];
  __shared__ float oc[kC];
  __shared__ int   cnt_s;

  const int i   = blockIdx.x;
  const int tid = threadIdx.x;
  if (tid == 0) cnt_s = 0;
  __syncthreads();

  const float* arow = adj + (size_t)i * kN;
  __builtin_prefetch(arow, 0, 1);
  for (int j = tid; j < kN; j += 256) {
    if (arow[j] > 0.f) {
      const int p = atomicAdd(&cnt_s, 1);
      idxs[p] = j;
    }
  }
  __syncthreads();
  const int cnt = cnt_s;

  const float t1i = t1[i];
  float lmax = -3.0e38f;
  for (int p = tid; p < cnt; p += 256) {
    float e = t1i + t2[idxs[p]];
    e = e > 0.f ? e : kAlpha * e;
    wv[p] = e;
    lmax  = fmaxf(lmax, e);
  }
  const float mx = block_max(lmax, red);

  float lsum = 0.f;
  for (int p = tid; p < cnt; p += 256) {
    const float w = __expf(wv[p] - mx);
    wv[p] = w;
    lsum += w;
  }
  const float invZ = 1.f / block_sum(lsum, red);

  if (tid < kC) {
    float acc = 0.f;
    for (int p = 0; p < cnt; ++p)
      acc += wv[p] * Wh2[(size_t)idxs[p] * kC + tid];
    acc *= invZ;
    oc[tid] = acc > 0.f ? acc : (__expf(acc) - 1.f);  // ELU
  }
  __syncthreads();

  if (tid < kC) {
    float m2 = oc[0];
    for (int c = 1; c < kC; ++c) m2 = fmaxf(m2, oc[c]);
    float s = 0.f;
    for (int c = 0; c < kC; ++c) s += __expf(oc[c] - m2);
    out[(size_t)i * kC + tid] = oc[tid] - m2 - __logf(s);
  }
}

// ---------------------------------------------------------------------------
extern "C" void kernel_launch(void* const* d_in, const int* in_sizes, int n_in,
                              void* d_out, int out_size, void* d_ws, size_t ws_size,
                              hipStream_t stream) {
  (void)in_sizes; (void)n_in; (void)out_size; (void)ws_size;
  const float* x       = (const float*)d_in[0];  // (4096, 512)
  const float* adj     = (const float*)d_in[1];  // (4096, 4096)
  const float* W_heads = (const float*)d_in[2];  // (8, 512, 64)
  const float* a_heads = (const float*)d_in[3];  // (8, 128, 1)
  const float* W_out   = (const float*)d_in[4];  // (512, 41)
  const float* a_out   = (const float*)d_in[5];  // (82, 1)
  float*       out     = (float*)d_out;          // (4096, 41)

  float* ws = (float*)d_ws;
  float* Wh   = ws;                                   // 8*4096*64   = 2,097,152
  float* s1   = Wh   + (size_t)kHeads * kN * kD;      // 32,768
  float* s2   = s1   + (size_t)kHeads * kN;           // 32,768
  float* hcat = s2   + (size_t)kHeads * kN;           // 4096*512    = 2,097,152
  float* Wh2  = hcat + (size_t)kN * kHeads * kD;      // 4096*41     = 167,936
  float* t1   = Wh2  + (size_t)kN * kC;               // 4,096
  float* t2   = t1   + (size_t)kN;                    // 4,096
  // total ~4.44M floats (~17.7 MB) of workspace

  k_gemm_heads<<<dim3(kN / 16, kHeads), 32, 0, stream>>>(x, W_heads, Wh);
  k_svec<<<(kHeads * kN + 255) / 256, 256, 0, stream>>>(Wh, a_heads, s1, s2);
  k_attn1<<<kN, 256, 0, stream>>>(adj, Wh, s1, s2, hcat);
  k_gemm_out<<<kN / 16, 32, 0, stream>>>(hcat, W_out, Wh2);
  k_tvec<<<(kN + 255) / 256, 256, 0, stream>>>(Wh2, a_out, t1, t2);
  k_attn2<<<kN, 256, 0, stream>>>(adj, Wh2, t1, t2, out);
}